// NTM_81106162418129
// MI455X (gfx1250) — compile-verified
//
#include <hip/hip_runtime.h>
#include <hip/hip_bf16.h>
#include <math.h>

typedef __attribute__((ext_vector_type(2))) float v2f;
typedef __attribute__((ext_vector_type(8))) float v8f;

static constexpr int    NN  = 262144;   // memory rows
static constexpr int    MM  = 128;      // memory width
static constexpr int    NH  = 1024;
static constexpr int    ZD  = 522;      // 2*(M+5)+2*M

// ---- workspace layout (float offsets) ----
static constexpr size_t WS_GATES = 0;                 // 4096
static constexpr size_t WS_V     = 4096;              // 128
static constexpr size_t WS_ZETA  = 4224;              // 522
static constexpr size_t WS_CONS  = 4800;              // 32 scalars
static constexpr size_t WS_Y     = 4832;              // 128
static constexpr size_t WS_READ  = 4960;              // 128
static constexpr size_t WS_P1R   = 8192;              // 1024
static constexpr size_t WS_P1W   = 12288;             // 1024
static constexpr size_t WS_P2R   = 16384;             // 1024
static constexpr size_t WS_P2W   = 17408;             // 1024
static constexpr size_t WS_PREAD = 18432;             // 256*128
static constexpr size_t WS_ER    = 65536;             // NN
static constexpr size_t WS_EW    = WS_ER  + NN;
static constexpr size_t WS_WGR   = WS_EW  + NN;
static constexpr size_t WS_WGW   = WS_WGR + NN;
static constexpr size_t WS_WPR   = WS_WGW + NN;
static constexpr size_t WS_WPW   = WS_WPR + NN;

// cons indices: 0 |k_r|, 1 |k_w|, 2 g_r, 3 g_w, 4-6 s_r, 7-9 s_w,
//               10 gamma_r, 11 gamma_w, 12 S_r, 13 S_w, 14 T_r, 15 T_w

// ============================================================================
// Generic GEMV via V_WMMA_F32_16X16X4_F32: out[r] = sum_k W[r,k]*vec[k] (+bias,+addend)
// One wave computes 16 output rows.  A = 16x4 tile of W, B column 0 = vec chunk.
// A layout (ISA 7.12.2): lanes 0-15 M=0..15, VGPR0=K0/K2(hi half), VGPR1=K1/K3.
// C layout: VGPR r, lanes0-15 -> M=r,N=lane ; lanes16-31 -> M=r+8,N=lane-16.
// ============================================================================
__global__ void k_gemv_wmma(const float* __restrict__ W, const float* __restrict__ vec,
                            const float* __restrict__ bias, const float* __restrict__ addend,
                            float* __restrict__ out, int rows, int K) {
    const int wave = threadIdx.x >> 5;
    const int lane = threadIdx.x & 31;
    const int tile = blockIdx.x * (blockDim.x >> 5) + wave;
    const int row0 = tile * 16;
    if (row0 >= rows) return;                      // wave-uniform exit
    const int m  = lane & 15;
    const int kh = lane >> 4;                      // 0: K0/K1, 1: K2/K3
    const int r  = row0 + m;
    const bool rok = (r < rows);
    const int  rc  = rok ? r : (rows - 1);
    const float* Wrow = W + (size_t)rc * K;

    v8f c = {0.f, 0.f, 0.f, 0.f, 0.f, 0.f, 0.f, 0.f};
#pragma unroll 4
    for (int k0 = 0; k0 < K; k0 += 4) {
        const int ka = k0 + 2 * kh;
        v2f a, b;
        a.x = rok ? Wrow[ka]     : 0.f;
        a.y = rok ? Wrow[ka + 1] : 0.f;
        b.x = (m == 0) ? vec[ka]     : 0.f;        // B column N=0 carries the vector
        b.y = (m == 0) ? vec[ka + 1] : 0.f;
        c = __builtin_amdgcn_wmma_f32_16x16x4_f32(false, a, false, b,
                                                  (short)0, c, false, false);
    }
    if (m == 0) {                                  // lanes 0 and 16 hold column N=0
        const int base = row0 + (kh ? 8 : 0);
#pragma unroll
        for (int i = 0; i < 8; ++i) {
            const int rr = base + i;
            if (rr < rows) {
                float val = c[i];
                if (bias)   val += bias[rr];
                if (addend) val += addend[rr];
                out[rr] = val;
            }
        }
    }
}

// ============================================================================
// LSTM elementwise: c_new = sig(f)*c + sig(i)*tanh(g); h_new = sig(o)*tanh(c_new)
// ============================================================================
__global__ void k_lstm(const float* __restrict__ gates, const float* __restrict__ c_old,
                       float* __restrict__ h_new, float* __restrict__ c_new) {
    int i = blockIdx.x * blockDim.x + threadIdx.x;
    if (i >= NH) return;
    float ig = gates[i], fg = gates[NH + i], gg = gates[2 * NH + i], og = gates[3 * NH + i];
    float si = 1.f / (1.f + expf(-ig));
    float sf = 1.f / (1.f + expf(-fg));
    float so = 1.f / (1.f + expf(-og));
    float cn = sf * c_old[i] + si * tanhf(gg);
    c_new[i] = cn;
    h_new[i] = so * tanhf(cn);
}

// ============================================================================
// Head setup: key norms, g, shift softmax, gamma (write-head gamma = zeta[521],
// replicating the reference's head_params[-1] on the full 389-long slice).
// ============================================================================
__global__ void k_head_setup(const float* __restrict__ zeta, float* __restrict__ cons) {
    __shared__ float sr[128], sw[128];
    int j = threadIdx.x;
    float kr = zeta[j] + 1e-16f;
    float kw = zeta[133 + j] + 1e-16f;
    sr[j] = kr * kr; sw[j] = kw * kw;
    __syncthreads();
    for (int s = 64; s > 0; s >>= 1) {
        if (j < s) { sr[j] += sr[j + s]; sw[j] += sw[j + s]; }
        __syncthreads();
    }
    if (j == 0) {
        cons[0] = sqrtf(sr[0]);
        cons[1] = sqrtf(sw[0]);
        cons[2] = 1.f / (1.f + expf(-zeta[128]));   // g_r
        cons[3] = 1.f / (1.f + expf(-zeta[261]));   // g_w
        {
            float a0 = zeta[129], a1 = zeta[130], a2 = zeta[131];
            float mx = fmaxf(a0, fmaxf(a1, a2));
            float e0 = expf(a0 - mx), e1 = expf(a1 - mx), e2 = expf(a2 - mx);
            float inv = 1.f / (e0 + e1 + e2);
            cons[4] = e0 * inv; cons[5] = e1 * inv; cons[6] = e2 * inv;
        }
        {
            float a0 = zeta[262], a1 = zeta[263], a2 = zeta[264];
            float mx = fmaxf(a0, fmaxf(a1, a2));
            float e0 = expf(a0 - mx), e1 = expf(a1 - mx), e2 = expf(a2 - mx);
            float inv = 1.f / (e0 + e1 + e2);
            cons[7] = e0 * inv; cons[8] = e1 * inv; cons[9] = e2 * inv;
        }
        cons[10] = 1.f + log1pf(expf(zeta[132]));   // gamma_r
        cons[11] = 1.f + log1pf(expf(zeta[521]));   // gamma_w (reference quirk)
    }
}

// ============================================================================
// Fused two-head cosine-similarity scan over the 128MB memory matrix.
// Thread-per-row: each lane privately accumulates dot_r/dot_w/||row||^2 over
// 32 float4 loads -> zero cross-lane shuffles in the hot loop (the wave-per-row
// variant spent ~15 shuffle ops per 512B row, which risks capping below HBM
// rate).  Keys broadcast from LDS.  e_r/e_w stores fully coalesced.
// ============================================================================
__global__ void k_sim(const float* __restrict__ memory, const float* __restrict__ zeta,
                      const float* __restrict__ cons,
                      float* __restrict__ e_r, float* __restrict__ e_w,
                      float* __restrict__ p1r, float* __restrict__ p1w) {
    __shared__ __attribute__((aligned(16))) float skr[128];
    __shared__ __attribute__((aligned(16))) float skw[128];
    __shared__ float red1[256], red2[256];
    const int tid = threadIdx.x;
    if (tid < 128) {
        skr[tid] = zeta[tid] + 1e-16f;
        skw[tid] = zeta[133 + tid] + 1e-16f;
    }
    __syncthreads();
    const float nkr = cons[0], nkw = cons[1];
    const size_t n = (size_t)blockIdx.x * 256 + tid;
    const float4* row = (const float4*)(memory + n * MM);
    const float4* kr4 = (const float4*)skr;     // same addr across lanes -> ds broadcast
    const float4* kw4 = (const float4*)skw;
    float dr = 0.f, dw = 0.f, ss = 0.f;
#pragma unroll 8
    for (int k = 0; k < 32; ++k) {
        float4 mv = row[k];
        float4 kr = kr4[k];
        float4 kw = kw4[k];
        float m0 = mv.x + 1e-16f, m1 = mv.y + 1e-16f;
        float m2 = mv.z + 1e-16f, m3 = mv.w + 1e-16f;
        dr = fmaf(m0, kr.x, dr); dr = fmaf(m1, kr.y, dr);
        dr = fmaf(m2, kr.z, dr); dr = fmaf(m3, kr.w, dr);
        dw = fmaf(m0, kw.x, dw); dw = fmaf(m1, kw.y, dw);
        dw = fmaf(m2, kw.z, dw); dw = fmaf(m3, kw.w, dw);
        ss = fmaf(m0, m0, ss);   ss = fmaf(m1, m1, ss);
        ss = fmaf(m2, m2, ss);   ss = fmaf(m3, m3, ss);
    }
    const float nm = sqrtf(ss);
    const float er = expf(dr / fmaxf(nm * nkr, 1e-8f));
    const float ew = expf(dw / fmaxf(nm * nkw, 1e-8f));
    e_r[n] = er;
    e_w[n] = ew;
    red1[tid] = er; red2[tid] = ew;
    __syncthreads();
    for (int s = 128; s > 0; s >>= 1) {
        if (tid < s) { red1[tid] += red1[tid + s]; red2[tid] += red2[tid + s]; }
        __syncthreads();
    }
    if (tid == 0) { p1r[blockIdx.x] = red1[0]; p1w[blockIdx.x] = red2[0]; }
}

// single-block deterministic reduce of two partial arrays
__global__ void k_reduce2(const float* __restrict__ pa, const float* __restrict__ pb,
                          int n, float* __restrict__ outA, float* __restrict__ outB) {
    __shared__ float sa[256], sb[256];
    float a = 0.f, b = 0.f;
    for (int i = threadIdx.x; i < n; i += 256) { a += pa[i]; b += pb[i]; }
    sa[threadIdx.x] = a; sb[threadIdx.x] = b;
    __syncthreads();
    for (int s = 128; s > 0; s >>= 1) {
        if (threadIdx.x < s) { sa[threadIdx.x] += sa[threadIdx.x + s]; sb[threadIdx.x] += sb[threadIdx.x + s]; }
        __syncthreads();
    }
    if (threadIdx.x == 0) { *outA = sa[0]; *outB = sb[0]; }
}

// w_g = g * softmax(sim) + (1-g) * w_prev   (both heads)
__global__ void k_wg(const float* __restrict__ e_r, const float* __restrict__ e_w,
                     const float* __restrict__ rw_prev, const float* __restrict__ ww_prev,
                     const float* __restrict__ cons,
                     float* __restrict__ wg_r, float* __restrict__ wg_w) {
    int i = blockIdx.x * blockDim.x + threadIdx.x;
    if (i >= NN) return;
    const float gr = cons[2], gw = cons[3];
    const float iSr = 1.f / cons[12], iSw = 1.f / cons[13];
    wg_r[i] = gr * e_r[i] * iSr + (1.f - gr) * rw_prev[i];
    wg_w[i] = gw * e_w[i] * iSw + (1.f - gw) * ww_prev[i];
}

// w_tild (circular shift convolution) then sharpen ^gamma; block partial sums
__global__ void k_wp(const float* __restrict__ wg_r, const float* __restrict__ wg_w,
                     const float* __restrict__ cons,
                     float* __restrict__ wp_r, float* __restrict__ wp_w,
                     float* __restrict__ p2r, float* __restrict__ p2w) {
    __shared__ float s1[256], s2[256];
    int i  = blockIdx.x * blockDim.x + threadIdx.x;
    int i1 = (i + 1) & (NN - 1);
    int i2 = (i + 2) & (NN - 1);
    float wtr = cons[4] * wg_r[i] + cons[5] * wg_r[i1] + cons[6] * wg_r[i2];
    float wtw = cons[7] * wg_w[i] + cons[8] * wg_w[i1] + cons[9] * wg_w[i2];
    float pr = powf(wtr, cons[10]);
    float pw = powf(wtw, cons[11]);
    wp_r[i] = pr; wp_w[i] = pw;
    s1[threadIdx.x] = pr; s2[threadIdx.x] = pw;
    __syncthreads();
    for (int s = 128; s > 0; s >>= 1) {
        if (threadIdx.x < s) { s1[threadIdx.x] += s1[threadIdx.x + s]; s2[threadIdx.x] += s2[threadIdx.x + s]; }
        __syncthreads();
    }
    if (threadIdx.x == 0) { p2r[blockIdx.x] = s1[0]; p2w[blockIdx.x] = s2[0]; }
}

// ============================================================================
// Fused: mem_new = memory*(1 - ww⊗erase) + ww⊗add  AND  partial of rw@mem_new.
// 256 blocks x 1024 rows; wave owns a 128-row slice, lane owns 4 columns.
// The per-row scalars wp_r/wp_w for the whole block (2x4KB) are staged into LDS
// with GLOBAL_LOAD_ASYNC_TO_LDS_B128 (GVS mode) + s_wait_asynccnt, replacing
// per-wave broadcast global loads with ds broadcasts.
// ============================================================================
__global__ void k_memupdate(const float* __restrict__ memory,
                            const float* __restrict__ wp_r, const float* __restrict__ wp_w,
                            const float* __restrict__ cons, const float* __restrict__ zeta,
                            float* __restrict__ mem_new, float* __restrict__ pread) {
    __shared__ __attribute__((aligned(16))) float s_wpr[1024];
    __shared__ __attribute__((aligned(16))) float s_wpw[1024];
    __shared__ float lds[8 * 128];
    const int wave = threadIdx.x >> 5;
    const int lane = threadIdx.x & 31;

    // --- async stage of this block's wp chunks into LDS (one b128 per thread each)
    {
        const float* gr = wp_r + (size_t)blockIdx.x * 1024;
        const float* gw = wp_w + (size_t)blockIdx.x * 1024;
        unsigned voff    = (unsigned)threadIdx.x * 16u;              // 256*16B = 4KB
        unsigned ldsoffr = (unsigned)(size_t)(&s_wpr[0]) + voff;     // flat LDS addr[31:0] = LDS offset
        unsigned ldsoffw = (unsigned)(size_t)(&s_wpw[0]) + voff;
        asm volatile("global_load_async_to_lds_b128 %0, %1, %2"
                     :: "v"(ldsoffr), "v"(voff), "s"(gr) : "memory");
        asm volatile("global_load_async_to_lds_b128 %0, %1, %2"
                     :: "v"(ldsoffw), "v"(voff), "s"(gw) : "memory");
        asm volatile("s_wait_asynccnt 0x0" ::: "memory");
    }
    __syncthreads();

    const float iTr = 1.f / (cons[14] + 1e-16f);
    const float iTw = 1.f / (cons[15] + 1e-16f);
    const float* er = zeta + 266;   // erase (not 16B aligned -> scalar loads)
    const float* ad = zeta + 394;   // add
    float e0 = er[lane * 4 + 0], e1 = er[lane * 4 + 1], e2 = er[lane * 4 + 2], e3 = er[lane * 4 + 3];
    float a0 = ad[lane * 4 + 0], a1 = ad[lane * 4 + 1], a2 = ad[lane * 4 + 2], a3 = ad[lane * 4 + 3];
    float4 acc = make_float4(0.f, 0.f, 0.f, 0.f);
    const size_t base = (size_t)blockIdx.x * 1024 + (size_t)wave * 128;
    for (int i = 0; i < 128; ++i) {
        const size_t n  = base + i;
        const size_t pn = (n + 4 < (size_t)NN) ? (n + 4) : (size_t)(NN - 1);
        __builtin_prefetch(memory + pn * MM + lane * 4, 0, 0);
        const float ww = s_wpw[wave * 128 + i] * iTw;   // ds broadcast
        const float rw = s_wpr[wave * 128 + i] * iTr;
        float4 mv = ((const float4*)(memory + n * MM))[lane];
        float4 mn;
        mn.x = mv.x * (1.f - ww * e0) + ww * a0;
        mn.y = mv.y * (1.f - ww * e1) + ww * a1;
        mn.z = mv.z * (1.f - ww * e2) + ww * a2;
        mn.w = mv.w * (1.f - ww * e3) + ww * a3;
        ((float4*)(mem_new + n * MM))[lane] = mn;
        acc.x += rw * mn.x; acc.y += rw * mn.y; acc.z += rw * mn.z; acc.w += rw * mn.w;
    }
    lds[wave * 128 + lane * 4 + 0] = acc.x;
    lds[wave * 128 + lane * 4 + 1] = acc.y;
    lds[wave * 128 + lane * 4 + 2] = acc.z;
    lds[wave * 128 + lane * 4 + 3] = acc.w;
    __syncthreads();
    if (threadIdx.x < 128) {
        float s = 0.f;
#pragma unroll
        for (int w = 0; w < 8; ++w) s += lds[w * 128 + threadIdx.x];
        pread[(size_t)blockIdx.x * 128 + threadIdx.x] = s;
    }
}

__global__ void k_reduce_read(const float* __restrict__ pread, int nblocks,
                              float* __restrict__ readv) {
    int j = threadIdx.x;   // 128 threads
    float s = 0.f;
    for (int b = 0; b < nblocks; ++b) s += pread[(size_t)b * 128 + j];
    readv[j] = s;
}

__global__ void k_logsoftmax(const float* __restrict__ y, float* __restrict__ out) {
    __shared__ float red[128];
    int j = threadIdx.x;
    float v = y[j];
    red[j] = v; __syncthreads();
    for (int s = 64; s > 0; s >>= 1) { if (j < s) red[j] = fmaxf(red[j], red[j + s]); __syncthreads(); }
    float mx = red[0]; __syncthreads();
    red[j] = expf(v - mx); __syncthreads();
    for (int s = 64; s > 0; s >>= 1) { if (j < s) red[j] += red[j + s]; __syncthreads(); }
    out[j] = v - (mx + logf(red[0]));
}

// ============================================================================
extern "C" void kernel_launch(void* const* d_in, const int* in_sizes, int n_in,
                              void* d_out, int out_size, void* d_ws, size_t ws_size,
                              hipStream_t stream) {
    const float* x       = (const float*)d_in[0];
    const float* h       = (const float*)d_in[1];
    const float* c_in    = (const float*)d_in[2];
    const float* rw_prev = (const float*)d_in[3];
    const float* ww_prev = (const float*)d_in[4];
    const float* memory  = (const float*)d_in[5];
    const float* W_ih    = (const float*)d_in[6];
    const float* W_hh    = (const float*)d_in[7];
    const float* b_ih    = (const float*)d_in[8];
    const float* b_hh    = (const float*)d_in[9];
    const float* W_v     = (const float*)d_in[10];
    const float* b_v     = (const float*)d_in[11];
    const float* W_zeta  = (const float*)d_in[12];
    const float* b_zeta  = (const float*)d_in[13];
    const float* W_read  = (const float*)d_in[14];
    const float* b_read  = (const float*)d_in[15];

    float* out    = (float*)d_out;
    float* h_new  = out;                               // 1024
    float* c_new  = out + 1024;                        // 1024
    float* memN   = out + 2048;                        // NN*MM
    float* out_ls = out + 2048 + (size_t)NN * MM;      // 128

    float* ws    = (float*)d_ws;
    float* gates = ws + WS_GATES;
    float* vws   = ws + WS_V;
    float* zws   = ws + WS_ZETA;
    float* cons  = ws + WS_CONS;
    float* yws   = ws + WS_Y;
    float* readv = ws + WS_READ;
    float* p1r   = ws + WS_P1R;
    float* p1w   = ws + WS_P1W;
    float* p2r   = ws + WS_P2R;
    float* p2w   = ws + WS_P2W;
    float* pread = ws + WS_PREAD;
    float* e_r   = ws + WS_ER;
    float* e_w   = ws + WS_EW;
    float* wg_r  = ws + WS_WGR;
    float* wg_w  = ws + WS_WGW;
    float* wp_r  = ws + WS_WPR;
    float* wp_w  = ws + WS_WPW;

    // 1) LSTM gates via WMMA (two accumulating GEMVs): 256 tiles of 16 rows
    k_gemv_wmma<<<64, 128, 0, stream>>>(W_ih, x, b_ih, nullptr, gates, 4 * NH, 128);
    k_gemv_wmma<<<64, 128, 0, stream>>>(W_hh, h, b_hh, gates,  gates, 4 * NH, NH);
    // 2) elementwise LSTM -> outputs 0/1
    k_lstm<<<4, 256, 0, stream>>>(gates, c_in, h_new, c_new);
    // 3) v and zeta via WMMA
    k_gemv_wmma<<<2, 128, 0, stream>>>(W_v,    c_new, b_v,    nullptr, vws, 128, NH);
    k_gemv_wmma<<<9, 128, 0, stream>>>(W_zeta, c_new, b_zeta, nullptr, zws, ZD,  NH);
    // 4) head scalars
    k_head_setup<<<1, 128, 0, stream>>>(zws, cons);
    // 5) fused two-head cosine scan (128MB, one pass, shuffle-free)
    k_sim<<<NN / 256, 256, 0, stream>>>(memory, zws, cons, e_r, e_w, p1r, p1w);
    k_reduce2<<<1, 256, 0, stream>>>(p1r, p1w, NN / 256, cons + 12, cons + 13);
    // 6) gated interpolation with previous weights
    k_wg<<<NN / 256, 256, 0, stream>>>(e_r, e_w, rw_prev, ww_prev, cons, wg_r, wg_w);
    // 7) circular shift + sharpen
    k_wp<<<NN / 256, 256, 0, stream>>>(wg_r, wg_w, cons, wp_r, wp_w, p2r, p2w);
    k_reduce2<<<1, 256, 0, stream>>>(p2r, p2w, NN / 256, cons + 14, cons + 15);
    // 8) fused memory erase/add update + read-vector partial (output 2)
    k_memupdate<<<256, 256, 0, stream>>>(memory, wp_r, wp_w, cons, zws, memN, pread);
    k_reduce_read<<<1, 128, 0, stream>>>(pread, 256, readv);
    // 9) y = v + read @ W_read.T + b_read via WMMA, then log_softmax (output 3)
    k_gemv_wmma<<<2, 128, 0, stream>>>(W_read, readv, b_read, vws, yws, 128, 128);
    k_logsoftmax<<<1, 128, 0, stream>>>(yws, out_ls);
}